// NCF_40157944217927
// MI455X (gfx1250) — compile-verified
//
#include <hip/hip_runtime.h>
#include <math.h>

typedef float v2f __attribute__((ext_vector_type(2)));
typedef float v8f __attribute__((ext_vector_type(8)));

namespace {
constexpr int ROWS = 64;    // rows (batch elements) per block
constexpr int TPB  = 128;   // 4 waves (wave32)
constexpr int CSTR = 132;   // LDS row stride for 128-wide activations (+4 pad)
constexpr int HSTR = 68;    // LDS row stride for 64-wide activations (+4 pad)
constexpr float BN_EPS = 1e-5f;
}

__global__ __launch_bounds__(TPB)
void ncf_fused_kernel(const float* __restrict__ dEmb,
                      const float* __restrict__ pEmb,
                      const int*   __restrict__ x,      // (B,2) int32
                      const float* __restrict__ W1,  const float* __restrict__ b1,
                      const float* __restrict__ g1,  const float* __restrict__ be1,
                      const float* __restrict__ mu1, const float* __restrict__ va1,
                      const float* __restrict__ W2,  const float* __restrict__ b2,
                      const float* __restrict__ g2,  const float* __restrict__ be2,
                      const float* __restrict__ mu2, const float* __restrict__ va2,
                      const float* __restrict__ Wfc, const float* __restrict__ bfc,
                      float* __restrict__ out)
{
    extern __shared__ float smem[];
    float* sC  = smem;                  // ROWS x CSTR : [d | p] concat
    float* sH1 = sC  + ROWS * CSTR;     // ROWS x CSTR : layer-1 activations
    float* sH2 = sH1 + ROWS * CSTR;     // ROWS x HSTR : layer-2 activations

    const int tid     = threadIdx.x;
    const int lane    = tid & 31;
    const int wave    = tid >> 5;
    const int rowBase = blockIdx.x * ROWS;

    // ---------------- gather: sC[r][0:64]=dEmb[xi], sC[r][64:128]=pEmb[xj]
    for (int i = tid; i < ROWS * 32; i += TPB) {
        const int r   = i >> 5;
        const int seg = i & 31;                 // 32 float4 segments per row
        const int c4  = (seg & 15) * 4;         // column within 64-wide half
        const bool isP = (seg >= 16);
        const long long idx = (long long)x[2 * (rowBase + r) + (isP ? 1 : 0)];
        const float* src = isP ? pEmb : dEmb;
        const float4 v = *(const float4*)(src + idx * 64 + c4);
        *(float4*)(sC + r * CSTR + c4 + (isP ? 64 : 0)) = v;
    }
    __syncthreads();

    // WMMA f32 16x16x4 lane mapping (ISA 7.12.2):
    //   A: lane L -> row m = L&15, holds K = {2*(L>>4), 2*(L>>4)+1}
    //   B: lane L -> col n = L&15, same K pair (mirrored)
    //   C/D: lane L -> col n = L&15, vgpr i -> row (L>>4)*8 + i
    const int m    = lane & 15;
    const int ko   = (lane >> 4) << 1;
    const int rOff = (lane >> 4) << 3;

    // ---------------- layer 1: H1 = tanh(bn1(sC @ W1 + b1)), N = 128
    for (int nT = wave; nT < 8; nT += 4) {
        const int n = nT * 16 + m;
        v8f c0 = {}, c1 = {}, c2 = {}, c3 = {};
        const float* wp = W1 + n;               // W1[k][n] row-major (128x128)
        #pragma unroll 8
        for (int k = 0; k < 128; k += 4) {
            v2f bf;
            bf.x = wp[(k + ko)     * 128];
            bf.y = wp[(k + ko + 1) * 128];
            const float* ap = sC + k + ko;
            v2f a0 = *(const v2f*)(ap + (m     ) * CSTR);
            v2f a1 = *(const v2f*)(ap + (m + 16) * CSTR);
            v2f a2 = *(const v2f*)(ap + (m + 32) * CSTR);
            v2f a3 = *(const v2f*)(ap + (m + 48) * CSTR);
            c0 = __builtin_amdgcn_wmma_f32_16x16x4_f32(false, a0, false, bf, (short)0, c0, false, false);
            c1 = __builtin_amdgcn_wmma_f32_16x16x4_f32(false, a1, false, bf, (short)0, c1, false, false);
            c2 = __builtin_amdgcn_wmma_f32_16x16x4_f32(false, a2, false, bf, (short)0, c2, false, false);
            c3 = __builtin_amdgcn_wmma_f32_16x16x4_f32(false, a3, false, bf, (short)0, c3, false, false);
        }
        const float sc = g1[n] * rsqrtf(va1[n] + BN_EPS);
        const float sh = b1[n] - mu1[n];
        const float bt = be1[n];
        #pragma unroll
        for (int i = 0; i < 8; ++i) {
            sH1[(rOff + i     ) * CSTR + n] = tanhf((c0[i] + sh) * sc + bt);
            sH1[(rOff + i + 16) * CSTR + n] = tanhf((c1[i] + sh) * sc + bt);
            sH1[(rOff + i + 32) * CSTR + n] = tanhf((c2[i] + sh) * sc + bt);
            sH1[(rOff + i + 48) * CSTR + n] = tanhf((c3[i] + sh) * sc + bt);
        }
    }
    __syncthreads();

    // ---------------- layer 2: H2 = tanh(bn2(H1 @ W2 + b2)), N = 64
    {
        const int n = wave * 16 + m;            // one N-tile per wave
        v8f c0 = {}, c1 = {}, c2 = {}, c3 = {};
        const float* wp = W2 + n;               // W2[k][n] row-major (128x64)
        #pragma unroll 8
        for (int k = 0; k < 128; k += 4) {
            v2f bf;
            bf.x = wp[(k + ko)     * 64];
            bf.y = wp[(k + ko + 1) * 64];
            const float* ap = sH1 + k + ko;
            v2f a0 = *(const v2f*)(ap + (m     ) * CSTR);
            v2f a1 = *(const v2f*)(ap + (m + 16) * CSTR);
            v2f a2 = *(const v2f*)(ap + (m + 32) * CSTR);
            v2f a3 = *(const v2f*)(ap + (m + 48) * CSTR);
            c0 = __builtin_amdgcn_wmma_f32_16x16x4_f32(false, a0, false, bf, (short)0, c0, false, false);
            c1 = __builtin_amdgcn_wmma_f32_16x16x4_f32(false, a1, false, bf, (short)0, c1, false, false);
            c2 = __builtin_amdgcn_wmma_f32_16x16x4_f32(false, a2, false, bf, (short)0, c2, false, false);
            c3 = __builtin_amdgcn_wmma_f32_16x16x4_f32(false, a3, false, bf, (short)0, c3, false, false);
        }
        const float sc = g2[n] * rsqrtf(va2[n] + BN_EPS);
        const float sh = b2[n] - mu2[n];
        const float bt = be2[n];
        #pragma unroll
        for (int i = 0; i < 8; ++i) {
            sH2[(rOff + i     ) * HSTR + n] = tanhf((c0[i] + sh) * sc + bt);
            sH2[(rOff + i + 16) * HSTR + n] = tanhf((c1[i] + sh) * sc + bt);
            sH2[(rOff + i + 32) * HSTR + n] = tanhf((c2[i] + sh) * sc + bt);
            sH2[(rOff + i + 48) * HSTR + n] = tanhf((c3[i] + sh) * sc + bt);
        }
    }
    __syncthreads();

    // ---------------- FC head: out = [gmf | H2] @ Wfc + bfc, N = 2
    {
        const int r = tid >> 1;                 // 128 threads -> 64 rows x 2 cols
        const int c = tid & 1;
        const float* cr = sC  + r * CSTR;
        const float* hr = sH2 + r * HSTR;
        float acc = bfc[c];
        #pragma unroll 8
        for (int k = 0; k < 64; ++k) {
            acc += (cr[k] * cr[64 + k]) * Wfc[k * 2 + c];   // gmf part
            acc += hr[k] * Wfc[(64 + k) * 2 + c];           // mlp part
        }
        out[(rowBase + r) * 2 + c] = acc;
    }
}

extern "C" void kernel_launch(void* const* d_in, const int* in_sizes, int n_in,
                              void* d_out, int out_size, void* d_ws, size_t ws_size,
                              hipStream_t stream) {
    const float* dEmb = (const float*)d_in[0];
    const float* pEmb = (const float*)d_in[1];
    const int*   x    = (const int*)  d_in[2];
    const float* W1   = (const float*)d_in[3];
    const float* b1   = (const float*)d_in[4];
    const float* g1   = (const float*)d_in[5];
    const float* be1  = (const float*)d_in[6];
    const float* mu1  = (const float*)d_in[7];
    const float* va1  = (const float*)d_in[8];
    const float* W2   = (const float*)d_in[9];
    const float* b2   = (const float*)d_in[10];
    const float* g2   = (const float*)d_in[11];
    const float* be2  = (const float*)d_in[12];
    const float* mu2  = (const float*)d_in[13];
    const float* va2  = (const float*)d_in[14];
    const float* Wfc  = (const float*)d_in[15];
    const float* bfc  = (const float*)d_in[16];
    float* out = (float*)d_out;

    const int batch   = in_sizes[2] / 2;        // 16384
    const int nblocks = batch / ROWS;           // 256
    const size_t shmem = (size_t)(2 * ROWS * CSTR + ROWS * HSTR) * sizeof(float); // ~85 KB

    hipLaunchKernelGGL(ncf_fused_kernel, dim3(nblocks), dim3(TPB), shmem, stream,
                       dEmb, pEmb, x, W1, b1, g1, be1, mu1, va1,
                       W2, b2, g2, be2, mu2, va2, Wfc, bfc, out);
}